// DGDUpdate_8589934763
// MI455X (gfx1250) — compile-verified
//
#include <hip/hip_runtime.h>

typedef __attribute__((ext_vector_type(2))) float v2f;
typedef __attribute__((ext_vector_type(4))) float v4f;
typedef __attribute__((ext_vector_type(8))) float v8f;

#define HEADS 256        // B*H
#define DDIM  512
#define STRIPS_PER_HEAD 32   // 512 rows / 16 rows per strip
#define WAVES_PER_BLOCK 8
#define THREADS (WAVES_PER_BLOCK * 32)
#define NBLOCKS ((HEADS * STRIPS_PER_HEAD) / WAVES_PER_BLOCK)  // 1024

// One wave32 per 16-row strip of one head's 512x512 matrix.
// Phase 1: pred[r0+j] for the 16 strip rows via chained V_WMMA_F32_16X16X4_F32:
//          A = k-chunk broadcast across rows, B = M^T chunk -> D[i][j] = pred[j]
//          replicated in every lane's accumulator (no cross-lane reduction).
// Phase 2: out = alpha*M + (eta*(pred - v))[i] * k[j], fully coalesced b128
//          (512B per load/store instruction), non-temporal stores.
__global__ __launch_bounds__(THREADS) void dgd_update_kernel(
    const float* __restrict__ M, const float* __restrict__ Kv,
    const float* __restrict__ Vv, const float* __restrict__ Al,
    const float* __restrict__ Et, float* __restrict__ O)
{
  const int wave  = threadIdx.x >> 5;
  const int lane  = threadIdx.x & 31;
  const int strip = blockIdx.x * WAVES_PER_BLOCK + wave;   // 0..8191
  const int head  = strip >> 5;                            // 0..255
  const int r0    = (strip & 31) << 4;                     // strip base row

  const float alpha = Al[head];
  const float eta   = Et[head];
  const float* kp = Kv + head * DDIM;
  const float* vp = Vv + head * DDIM;
  const float* Mh = M + (size_t)head * DDIM * DDIM;
  float*       Oh = O + (size_t)head * DDIM * DDIM;

  const int half = lane >> 4;   // K-pair selector per ISA 16x4 f32 A/B layout
  const int lm   = lane & 15;

  // ---------------- Phase 1: prediction via WMMA ----------------
  // B[r][j] = M[r0+j][4t+r]  -> lane loads b64 from its row at col 4t+2*half
  // A[i][r] = k[4t+r] (all i) -> lane loads b64 from k at col 4t+2*half
  const float* Mrow = Mh + (size_t)(r0 + lm) * DDIM + 2 * half;
  const float* kq   = kp + 2 * half;

  v8f acc = {0.f, 0.f, 0.f, 0.f, 0.f, 0.f, 0.f, 0.f};
#pragma unroll 8
  for (int t = 0; t < DDIM / 4; ++t) {
    v2f a = *(const v2f*)(kq   + 4 * t);
    v2f b = *(const v2f*)(Mrow + 4 * t);
    acc = __builtin_amdgcn_wmma_f32_16x16x4_f32(
        /*neg_a=*/false, a, /*neg_b=*/false, b,
        /*c_mod=*/(short)0, acc, /*reuse_a=*/false, /*reuse_b=*/false);
  }
  // Every element of acc holds pred[r0+lm]; fold in v and eta once.
  const float e = eta * (acc[0] - vp[r0 + lm]);

  // ---------------- Phase 2: streaming rank-1 update ----------------
  // k tiles for this lane's column slots (reused for all 16 rows)
  const v4f kv0 = *(const v4f*)(kp +   0 + 4 * lane);
  const v4f kv1 = *(const v4f*)(kp + 128 + 4 * lane);
  const v4f kv2 = *(const v4f*)(kp + 256 + 4 * lane);
  const v4f kv3 = *(const v4f*)(kp + 384 + 4 * lane);

#pragma unroll 4
  for (int i = 0; i < 16; ++i) {
    const float ei = __shfl(e, i, 32);   // e for strip row i (lanes 0-15 hold it)
    const float* Mi = Mh + (size_t)(r0 + i) * DDIM + 4 * lane;
    float*       Oi = Oh + (size_t)(r0 + i) * DDIM + 4 * lane;
    v4f m0 = *(const v4f*)(Mi +   0);
    v4f m1 = *(const v4f*)(Mi + 128);
    v4f m2 = *(const v4f*)(Mi + 256);
    v4f m3 = *(const v4f*)(Mi + 384);
    __builtin_nontemporal_store(alpha * m0 + ei * kv0, (v4f*)(Oi +   0));
    __builtin_nontemporal_store(alpha * m1 + ei * kv1, (v4f*)(Oi + 128));
    __builtin_nontemporal_store(alpha * m2 + ei * kv2, (v4f*)(Oi + 256));
    __builtin_nontemporal_store(alpha * m3 + ei * kv3, (v4f*)(Oi + 384));
  }
}

extern "C" void kernel_launch(void* const* d_in, const int* in_sizes, int n_in,
                              void* d_out, int out_size, void* d_ws, size_t ws_size,
                              hipStream_t stream) {
  const float* M  = (const float*)d_in[0];
  const float* K  = (const float*)d_in[1];
  const float* V  = (const float*)d_in[2];
  const float* A  = (const float*)d_in[3];
  const float* E  = (const float*)d_in[4];
  float*       O  = (float*)d_out;
  (void)in_sizes; (void)n_in; (void)out_size; (void)d_ws; (void)ws_size;
  hipLaunchKernelGGL(dgd_update_kernel, dim3(NBLOCKS), dim3(THREADS), 0, stream,
                     M, K, V, A, E, O);
}